// FeedForwardLoudnessControll_19009525252269
// MI455X (gfx1250) — compile-verified
//
#include <hip/hip_runtime.h>
#include <math.h>

#define HOP 256
#define WIN 24              // K-1 history segments

typedef __attribute__((ext_vector_type(2))) float v2f;
typedef __attribute__((ext_vector_type(8))) float v8f;

// Constants (log2-based so transcendentals lower to v_exp_f32 / v_log_f32)
#define RLOG2        ((float)(0.16 * 3.3219280948873623 / 2550.0)) // per-sample weight rate, log2 domain
#define INV_L        (1.0f / 6400.0f)
#define EPS_F        (1e-10f)
#define TEN_LOG10_2  (3.0102999566398120f)   // 10*log10(2): loudness = log2(e)*this
#define G2E_ENERGY   (0.33219280948873623f)  // 0.1 *log2(10): 10^(g/10) = exp2(g*this)
#define G2E_AMP      (0.16609640474436811f)  // 0.05*log2(10): 10^(g/20) = exp2(g*this)
#define GSTEP        (0.16f)                 // INC == DEC
#define THR_F        (0.5f)

// Hardware transcendentals: v_exp_f32 / v_log_f32 (both base-2)
__device__ __forceinline__ float fast_exp2(float v) {
#if __has_builtin(__builtin_amdgcn_exp2f)
  return __builtin_amdgcn_exp2f(v);
#else
  return exp2f(v);
#endif
}
__device__ __forceinline__ float fast_log2(float v) {
#if __has_builtin(__builtin_amdgcn_logf)
  return __builtin_amdgcn_logf(v);
#else
  return log2f(v);
#endif
}

// ---------------------------------------------------------------------------
// Phase 1: per-segment weighted energies E0/Ep/Em  ==  GEMM (nseg x 256)x(256 x 16)
// One wave per 16-segment tile, V_WMMA_F32_16X16X4_F32 accumulating over K=256.
// B columns: 0 -> 1.0, 1 -> 10^(+0.16*j/2550), 2 -> 10^(-0.16*j/2550), 3..15 -> 0
// ---------------------------------------------------------------------------
__global__ __launch_bounds__(32) void energies_kernel(const float* __restrict__ x,
                                                      float* __restrict__ E0,
                                                      float* __restrict__ Ep,
                                                      float* __restrict__ Em,
                                                      int n_seg) {
  const int lane = threadIdx.x;
  const int seg_base = blockIdx.x * 16;
  const int row  = lane & 15;
  const int half = lane >> 4;     // 0: K-slots {0,1}, 1: K-slots {2,3}
  const int koff = half * 2;
  const int seg  = seg_base + row;
  const bool valid = seg < n_seg;
  const float* rp = x + (size_t)(valid ? seg : 0) * HOP + koff;

#if __has_builtin(__builtin_amdgcn_wmma_f32_16x16x4_f32)
  const int col = row;  // B-fragment column index == lane&15
  const float rate = (col == 1) ? RLOG2 : (col == 2) ? -RLOG2 : 0.0f;
  const bool bact = (col < 3);
  float b0 = bact ? fast_exp2(rate * (float)koff)       : 0.0f;
  float b1 = bact ? fast_exp2(rate * (float)(koff + 1)) : 0.0f;
  const float bstep = fast_exp2(rate * 4.0f);  // advance weights by 4 K-steps per chunk

  v8f acc = {};
  #pragma unroll 4
  for (int c = 0; c < HOP / 4; ++c) {
    float2 xv = *(const float2*)(rp + 4 * c);
    v2f a; a.x = valid ? xv.x * xv.x : 0.0f;
           a.y = valid ? xv.y * xv.y : 0.0f;
    v2f b; b.x = b0; b.y = b1;
    // D(16x16,f32) += A(16x4,f32) x B(4x16,f32)  -- exact f32 accumulation
    acc = __builtin_amdgcn_wmma_f32_16x16x4_f32(false, a, false, b,
                                                (short)0, acc, false, false);
    b0 *= bstep; b1 *= bstep;
  }

  // C/D layout: VGPR r, lanes 0-15 -> (M=r, N=lane); lanes 16-31 -> (M=r+8, N=lane-16)
  const int mbase = half * 8;
  if (col < 3) {
    float* dst = (col == 0) ? E0 : (col == 1) ? Ep : Em;
    #pragma unroll
    for (int r = 0; r < 8; ++r) {
      int oseg = seg_base + mbase + r;
      if (oseg < n_seg) dst[oseg] = acc[r];
    }
  }
#else
  // Scalar fallback: lanes 0..15 each reduce one segment
  if (lane < 16 && valid) {
    const float* xr = x + (size_t)seg * HOP;
    float s0 = 0.f, sp = 0.f, sm = 0.f, wp = 1.f, wm = 1.f;
    const float up = fast_exp2(RLOG2), um = fast_exp2(-RLOG2);
    for (int j = 0; j < HOP; ++j) {
      float v = xr[j], v2 = v * v;
      s0 += v2; sp += v2 * wp; sm += v2 * wm;
      wp *= up; wm *= um;
    }
    E0[seg] = s0; Ep[seg] = sp; Em[seg] = sm;
  }
#endif
}

// ---------------------------------------------------------------------------
// Phase 2: sequential gain scan (latency-bound; minimized dependency chain)
// ---------------------------------------------------------------------------
__device__ __forceinline__ void agc_step(const float* __restrict__ E0,
                                         const float* __restrict__ Ep,
                                         const float* __restrict__ Em,
                                         float* __restrict__ gains,
                                         float& S, float& g, float& slot, int& t) {
  const int s = WIN + t;
  const float E0s = E0[s], Eps = Ep[s], Ems = Em[s];
  const float be   = S + E0s;                                     // block energy
  const float loud = fast_log2(be * INV_L + EPS_F) * TEN_LOG10_2; // 10*log10(be/L+eps)
  const float err  = loud + 23.0f + g;                            // loud - TARGET + g
  float gn, Esel;
  if (err < -THR_F)      { gn = g + GSTEP; Esel = Eps; }
  else if (err > THR_F)  { gn = g - GSTEP; Esel = Ems; }
  else                   { gn = g;         Esel = E0s; }
  const float en = fast_exp2(g * G2E_ENERGY) * Esel;              // 10^(g/10)*E_delta
  S = (S - slot) + en;                                            // running window sum
  slot = en;                                                      // ring buffer (reg-resident)
  g = gn;
  gains[t + 1] = gn;
  ++t;
}

__global__ __launch_bounds__(32) void scan_kernel(const float* __restrict__ E0,
                                                  const float* __restrict__ Ep,
                                                  const float* __restrict__ Em,
                                                  float* __restrict__ gains,
                                                  int n_scan) {
  if (threadIdx.x != 0) return;
  float win[WIN];
  float S = 0.0f;
  #pragma unroll
  for (int k = 0; k < WIN; ++k) { win[k] = E0[k]; S += win[k]; }
  float g = 0.0f;
  gains[0] = 0.0f;

  int t = 0;
  const int nfull = n_scan / WIN;   // 92136 = 24 * 3839 -> exact for this input
  for (int blk = 0; blk < nfull; ++blk) {
    __builtin_prefetch(&E0[WIN + t + 4 * WIN], 0, 0);
    __builtin_prefetch(&Ep[WIN + t + 4 * WIN], 0, 0);
    __builtin_prefetch(&Em[WIN + t + 4 * WIN], 0, 0);
    #pragma unroll
    for (int u = 0; u < WIN; ++u)   // slot index is compile-time -> ring stays in VGPRs
      agc_step(E0, Ep, Em, gains, S, g, win[u], t);
  }
  #pragma unroll
  for (int u = 0; u < WIN; ++u)     // remainder (<24), still compile-time slots
    if (t < n_scan) agc_step(E0, Ep, Em, gains, S, g, win[u], t);
}

// ---------------------------------------------------------------------------
// Phase 3: apply per-sample gain ramp + clip (pure bandwidth)
// ---------------------------------------------------------------------------
__global__ __launch_bounds__(256) void apply_kernel(const float* __restrict__ x,
                                                    const float* __restrict__ gains,
                                                    float* __restrict__ out,
                                                    int n_seg) {
  const int s = blockIdx.x;
  const int j = threadIdx.x;
  const size_t idx = (size_t)s * HOP + j;
  float v = x[idx];
  if (s >= WIN) {
    const int t = s - WIN;
    const float gsv  = gains[t];
    const float gnv  = gains[t + 1];
    const float gm   = gsv + (gnv - gsv) * ((float)j * (1.0f / 255.0f)); // linspace ramp
    v *= fast_exp2(gm * G2E_AMP);                                        // 10^(gm/20)
  }
  out[idx] = fminf(fmaxf(v, -1.0f), 1.0f);
}

// ---------------------------------------------------------------------------
extern "C" void kernel_launch(void* const* d_in, const int* in_sizes, int n_in,
                              void* d_out, int out_size, void* d_ws, size_t ws_size,
                              hipStream_t stream) {
  (void)n_in; (void)out_size; (void)ws_size;
  const float* x = (const float*)d_in[0];
  float* out = (float*)d_out;

  const int n      = in_sizes[0];
  const int n_seg  = n / HOP;        // 92160
  const int n_scan = n_seg - WIN;    // 92136

  float* E0    = (float*)d_ws;       // [n_seg]
  float* Ep    = E0 + n_seg;         // [n_seg]
  float* Em    = Ep + n_seg;         // [n_seg]
  float* gains = Em + n_seg;         // [n_scan + 1]   (~1.5 MB total workspace)

  const int tiles = (n_seg + 15) / 16;
  hipLaunchKernelGGL(energies_kernel, dim3(tiles), dim3(32), 0, stream, x, E0, Ep, Em, n_seg);
  hipLaunchKernelGGL(scan_kernel,     dim3(1),     dim3(32), 0, stream, E0, Ep, Em, gains, n_scan);
  hipLaunchKernelGGL(apply_kernel,    dim3(n_seg), dim3(256), 0, stream, x, gains, out, n_seg);
}